// AttentionDecoder_51866025067024
// MI455X (gfx1250) — compile-verified
//
#include <hip/hip_runtime.h>
#include <hip/hip_bf16.h>
#include <math.h>

#define Vv 50257
#define Hh 1024
#define Bb 16
#define Ll 100
#define Lp 112  // L padded to multiple of 16

typedef float v2f __attribute__((ext_vector_type(2)));
typedef float v8f __attribute__((ext_vector_type(8)));

// ---------------------------------------------------------------------------
// One K-segment of the concat-A GEMM, double-buffered.
// pA  : A segment base for this lane (already offset by m*K + kh)
// pB[t]: W row pointers for this lane (already offset by nc*K + kh)
// boff : element offset into the W rows where this segment starts
// Klen : segment length (multiple of 4, >= 4)
// W loads are non-temporal (weights are single-use streams).
// ---------------------------------------------------------------------------
template <int NT>
__device__ __forceinline__ void kseg_accum(const float* __restrict__ pA,
                                           const float* const* pB, int boff,
                                           int Klen, v8f* acc) {
    const int steps = Klen >> 2;
    v2f a = *(const v2f*)pA;
    v2f b[NT];
#pragma unroll
    for (int t = 0; t < NT; ++t)
        b[t] = __builtin_nontemporal_load((const v2f*)(pB[t] + boff));

#pragma unroll 2
    for (int s = 1; s < steps; ++s) {
        // prefetch next step's fragments before consuming current ones
        v2f an = *(const v2f*)(pA + 4 * s);
        v2f bn[NT];
#pragma unroll
        for (int t = 0; t < NT; ++t)
            bn[t] = __builtin_nontemporal_load(
                (const v2f*)(pB[t] + boff + 4 * s));
#pragma unroll
        for (int t = 0; t < NT; ++t)
            acc[t] = __builtin_amdgcn_wmma_f32_16x16x4_f32(
                false, a, false, b[t], (short)0, acc[t], false, false);
        a = an;
#pragma unroll
        for (int t = 0; t < NT; ++t) b[t] = bn[t];
    }
#pragma unroll
    for (int t = 0; t < NT; ++t)
        acc[t] = __builtin_amdgcn_wmma_f32_16x16x4_f32(
            false, a, false, b[t], (short)0, acc[t], false, false);
}

// ---------------------------------------------------------------------------
// WMMA f32 GEMM: out[16,N] = [A0 | A1] (16 x (K0+K1)) @ W[N,K]^T + bias.
// A0: [16,K0], A1: [16,K1] row-major, K0/K1 multiples of 4.
// W: [N,K] row-major. Each wave owns NT consecutive 16-wide N tiles,
// grid-stride over tile groups. V_WMMA_F32_16X16X4_F32, K in steps of 4.
// ---------------------------------------------------------------------------
template <int NT>
__global__ void gemm16_wmma_f32(const float* __restrict__ A0, int K0,
                                const float* __restrict__ A1, int K1,
                                const float* __restrict__ W,
                                const float* __restrict__ bias,
                                float* __restrict__ out, int ldout,
                                int N, int do_relu) {
    const int lane = threadIdx.x & 31;
    const int wave = threadIdx.x >> 5;
    const int m    = lane & 15;        // A row / output column-within-tile
    const int kh   = (lane >> 4) << 1; // 0 for lanes 0-15, 2 for lanes 16-31
    const int K    = K0 + K1;
    const int ntiles  = (N + 15) >> 4;
    const int ngroups = (ntiles + NT - 1) / NT;
    const int wpb     = blockDim.x >> 5;

    const float* pA0 = A0 ? (A0 + (size_t)m * K0 + kh) : nullptr;
    const float* pA1 = A1 ? (A1 + (size_t)m * K1 + kh) : nullptr;

    for (int g = blockIdx.x * wpb + wave; g < ngroups;
         g += gridDim.x * wpb) {
        const int n0 = g * NT * 16;
        const float* pB[NT];
#pragma unroll
        for (int t = 0; t < NT; ++t) {
            int n  = n0 + t * 16 + m;
            int nc = (n < N) ? n : (N - 1);
            pB[t] = W + (size_t)nc * K + kh;
        }

        v8f acc[NT];
#pragma unroll
        for (int t = 0; t < NT; ++t) acc[t] = (v8f){};

        if (K0 > 0) kseg_accum<NT>(pA0, pB, 0, K0, acc);
        if (K1 > 0) kseg_accum<NT>(pA1, pB, K0, K1, acc);

        const int mbase = (lane >> 4) << 3; // 0 or 8
#pragma unroll
        for (int t = 0; t < NT; ++t) {
            const int n = n0 + t * 16 + m;
            if (n < N) {
                const float bv = bias[n];
#pragma unroll
                for (int r = 0; r < 8; ++r) {
                    float v = acc[t][r] + bv;
                    if (do_relu) v = fmaxf(v, 0.0f);
                    out[(size_t)(mbase + r) * ldout + n] = v;
                }
            }
        }
    }
}

// ---------------------------------------------------------------------------
// embedded[b,h] = emb[inputs[b], h]
// ---------------------------------------------------------------------------
__global__ void embed_kernel(const int* __restrict__ inputs,
                             const float* __restrict__ emb,
                             float* __restrict__ out) {
    int i = blockIdx.x * blockDim.x + threadIdx.x; // 16*1024 threads
    int b = i >> 10, h = i & 1023;
    out[i] = emb[(size_t)inputs[b] * Hh + h];
}

// ---------------------------------------------------------------------------
// Softmax over L=100 per row; one wave per batch row. Writes ws copy + d_out.
// logits laid out [16, Lp].
// ---------------------------------------------------------------------------
__global__ void attn_softmax_kernel(const float* __restrict__ logits,
                                    float* __restrict__ w_ws,
                                    float* __restrict__ w_out) {
    const int lane = threadIdx.x & 31;
    const int b    = threadIdx.x >> 5; // 16 waves
    float vals[4];
    float mx = -3.4e38f;
#pragma unroll
    for (int j = 0; j < 4; ++j) {
        int l = lane + 32 * j;
        vals[j] = (l < Ll) ? logits[b * Lp + l] : -3.4e38f;
        mx = fmaxf(mx, vals[j]);
    }
#pragma unroll
    for (int o = 16; o > 0; o >>= 1) mx = fmaxf(mx, __shfl_xor(mx, o, 32));
    float s = 0.f;
#pragma unroll
    for (int j = 0; j < 4; ++j) {
        int l = lane + 32 * j;
        vals[j] = (l < Ll) ? __expf(vals[j] - mx) : 0.f;
        s += vals[j];
    }
#pragma unroll
    for (int o = 16; o > 0; o >>= 1) s += __shfl_xor(s, o, 32);
    const float inv = 1.0f / s;
#pragma unroll
    for (int j = 0; j < 4; ++j) {
        int l = lane + 32 * j;
        if (l < Ll) {
            float wv = vals[j] * inv;
            w_ws[b * Ll + l]  = wv;
            w_out[b * Ll + l] = wv;
        }
    }
}

// ---------------------------------------------------------------------------
// attn_applied[b,h] = sum_l w[b,l] * enc[l,b,h]
// ---------------------------------------------------------------------------
__global__ void attn_apply_kernel(const float* __restrict__ w,
                                  const float* __restrict__ enc,
                                  float* __restrict__ out) {
    int i = blockIdx.x * blockDim.x + threadIdx.x; // 16*1024
    int b = i >> 10, h = i & 1023;
    float s = 0.f;
    for (int l = 0; l < Ll; ++l)
        s += w[b * Ll + l] * enc[((size_t)l * Bb + b) * Hh + h];
    out[i] = s;
}

// ---------------------------------------------------------------------------
// GRU gates (torch order r,z,n): h1 = (1-z)*n + z*h0
// ---------------------------------------------------------------------------
__global__ void gru_kernel(const float* __restrict__ gi,
                           const float* __restrict__ gh,
                           const float* __restrict__ h0,
                           float* __restrict__ h1_ws,
                           float* __restrict__ h1_out) {
    int i = blockIdx.x * blockDim.x + threadIdx.x; // 16*1024
    int b = i >> 10, h = i & 1023;
    const float* gib = gi + (size_t)b * 3 * Hh;
    const float* ghb = gh + (size_t)b * 3 * Hh;
    float r = 1.0f / (1.0f + __expf(-(gib[h] + ghb[h])));
    float z = 1.0f / (1.0f + __expf(-(gib[Hh + h] + ghb[Hh + h])));
    float n = tanhf(gib[2 * Hh + h] + r * ghb[2 * Hh + h]);
    float hv = (1.0f - z) * n + z * h0[i];
    h1_ws[i]  = hv;
    h1_out[i] = hv;
}

// ---------------------------------------------------------------------------
// Per-row max and log-sum-exp over V (one block per batch row).
// ---------------------------------------------------------------------------
__global__ void rowstats_kernel(const float* __restrict__ logits,
                                float* __restrict__ stats) {
    __shared__ float sm[256];
    const int b = blockIdx.x, t = threadIdx.x;
    const float* row = logits + (size_t)b * Vv;
    float mx = -3.4e38f;
    for (int v = t; v < Vv; v += 256) mx = fmaxf(mx, row[v]);
    sm[t] = mx; __syncthreads();
    for (int o = 128; o > 0; o >>= 1) {
        if (t < o) sm[t] = fmaxf(sm[t], sm[t + o]);
        __syncthreads();
    }
    mx = sm[0]; __syncthreads();
    float s = 0.f;
    for (int v = t; v < Vv; v += 256) s += __expf(row[v] - mx);
    sm[t] = s; __syncthreads();
    for (int o = 128; o > 0; o >>= 1) {
        if (t < o) sm[t] += sm[t + o];
        __syncthreads();
    }
    if (t == 0) { stats[2 * b] = mx; stats[2 * b + 1] = logf(sm[0]); }
}

// ---------------------------------------------------------------------------
// In-place log-softmax finish: x -= (max + log(sumexp))
// ---------------------------------------------------------------------------
__global__ void logsoftmax_kernel(float* __restrict__ out,
                                  const float* __restrict__ stats) {
    size_t i = (size_t)blockIdx.x * blockDim.x + threadIdx.x;
    if (i >= (size_t)Bb * Vv) return;
    int b = (int)(i / Vv);
    out[i] = out[i] - stats[2 * b] - stats[2 * b + 1];
}

// ---------------------------------------------------------------------------
extern "C" void kernel_launch(void* const* d_in, const int* in_sizes, int n_in,
                              void* d_out, int out_size, void* d_ws, size_t ws_size,
                              hipStream_t stream) {
    const int*   inputs = (const int*)  d_in[0];
    const float* h0     = (const float*)d_in[1];  // [1,B,H] -> [B,H]
    const float* enc    = (const float*)d_in[2];  // [L,B,H]
    const float* emb    = (const float*)d_in[3];  // [V,H]
    const float* W_attn = (const float*)d_in[4];  // [L,2H]
    const float* b_attn = (const float*)d_in[5];
    const float* W_comb = (const float*)d_in[6];  // [H,2H]
    const float* b_comb = (const float*)d_in[7];
    const float* W_ih   = (const float*)d_in[8];  // [3H,H]
    const float* W_hh   = (const float*)d_in[9];  // [3H,H]
    const float* b_ih   = (const float*)d_in[10];
    const float* b_hh   = (const float*)d_in[11];
    const float* W_out  = (const float*)d_in[12]; // [V,H]
    const float* b_out  = (const float*)d_in[13];

    float* out      = (float*)d_out;
    float* out_h1   = out + (size_t)Bb * Vv;              // [B,H]
    float* out_attn = out_h1 + (size_t)Bb * Hh;           // [B,L]

    float* ws        = (float*)d_ws;
    float* ws_emb    = ws;                         // 16*1024
    float* ws_logits = ws_emb    + Bb * Hh;        // 16*Lp
    float* ws_attnw  = ws_logits + Bb * Lp;        // 16*L
    float* ws_appl   = ws_attnw  + Bb * Ll;        // 16*1024
    float* ws_x      = ws_appl   + Bb * Hh;        // 16*1024
    float* ws_gi     = ws_x      + Bb * Hh;        // 16*3072
    float* ws_gh     = ws_gi     + Bb * 3 * Hh;    // 16*3072
    float* ws_h1     = ws_gh     + Bb * 3 * Hh;    // 16*1024
    float* ws_stats  = ws_h1     + Bb * Hh;        // 32

    // 1. embedding gather
    embed_kernel<<<64, 256, 0, stream>>>(inputs, emb, ws_emb);

    // 2. attn_logits = [embedded | h0] @ W_attn^T + b_attn   (N=100, K=2048)
    gemm16_wmma_f32<1><<<1, 256, 0, stream>>>(ws_emb, Hh, h0, Hh, W_attn,
                                              b_attn, ws_logits, Lp, Ll, 0);

    // 3. softmax over L; write to ws and to d_out attention-weights region
    attn_softmax_kernel<<<1, 512, 0, stream>>>(ws_logits, ws_attnw, out_attn);

    // 4. attention_applied = einsum("bl,lbh->bh", w, enc)
    attn_apply_kernel<<<64, 256, 0, stream>>>(ws_attnw, enc, ws_appl);

    // 5. x = relu([embedded | attn_applied] @ W_comb^T + b_comb) (N=1024, K=2048)
    //    64 tiles -> 16 groups of 4
    gemm16_wmma_f32<4><<<2, 256, 0, stream>>>(ws_emb, Hh, ws_appl, Hh, W_comb,
                                              b_comb, ws_x, Hh, Hh, 1);

    // 6. gi = x @ W_ih^T + b_ih   (N=3072, K=1024): 192 tiles -> 48 groups
    gemm16_wmma_f32<4><<<6, 256, 0, stream>>>(ws_x, Hh, nullptr, 0, W_ih,
                                              b_ih, ws_gi, 3 * Hh, 3 * Hh, 0);

    // 7. gh = h0 @ W_hh^T + b_hh  (N=3072, K=1024)
    gemm16_wmma_f32<4><<<6, 256, 0, stream>>>(h0, Hh, nullptr, 0, W_hh,
                                              b_hh, ws_gh, 3 * Hh, 3 * Hh, 0);

    // 8. GRU gating -> h1 (ws + d_out h1 region)
    gru_kernel<<<64, 256, 0, stream>>>(ws_gi, ws_gh, h0, ws_h1, out_h1);

    // 9. logits = h1 @ W_out^T + b_out  (N=50257, K=1024) -> d_out directly.
    //    HBM-bound phase: NT=2 -> 1571 tile groups, 197*8 = 1576 waves
    //    streaming 206 MB of W_out with non-temporal loads.
    gemm16_wmma_f32<2><<<197, 256, 0, stream>>>(ws_h1, Hh, nullptr, 0, W_out,
                                                b_out, out, Vv, Vv, 0);

    // 10-11. log-softmax over V, in place on d_out.
    rowstats_kernel<<<16, 256, 0, stream>>>(out, ws_stats);
    logsoftmax_kernel<<<(Bb * Vv + 255) / 256, 256, 0, stream>>>(out, ws_stats);
}